// SkillPredictor_73899207295509
// MI455X (gfx1250) — compile-verified
//
#include <hip/hip_runtime.h>
#include <stdint.h>

#define T_TOKENS     65536      // B*N = 16*4096
#define DIM          64
#define CD           16
#define KCODES       500
#define KTILES       32         // ceil(500/16)
#define DECAY        0.99f
#define ONE_M_D      0.01f
#define COMMIT_W     0.25f
#define EPS_F        1e-5f
#define FUSED_BLOCKS 512        // 65536 tokens / (16 per wave * 8 waves)

typedef float v2f __attribute__((ext_vector_type(2)));
typedef float v8f __attribute__((ext_vector_type(8)));

__device__ __forceinline__ v8f wmma_f32(v2f a, v2f b, v8f c) {
    // V_WMMA_F32_16X16X4_F32: D = A(16x4) * B(4x16) + C(16x16), full f32
    return __builtin_amdgcn_wmma_f32_16x16x4_f32(false, a, false, b, (short)0, c,
                                                 false, false);
}

// ---------------------------------------------------------------------------
// Fused: z = x@W_in+b_in ; idx = argmin_k ||z-e_k||^2 ; out = e_idx@W_out+b_out
// Also stores z to workspace and per-block commitment-loss partials.
// One wave (32 lanes) owns a 16-token tile; 8 waves (256 thr) per block.
// ---------------------------------------------------------------------------
__global__ __launch_bounds__(256) void vq_fused_kernel(
    const float* __restrict__ x,
    const float* __restrict__ W_in,
    const float* __restrict__ b_in,
    const float* __restrict__ W_out,
    const float* __restrict__ b_out,
    const float* __restrict__ embed,
    float* __restrict__ out,
    float* __restrict__ idx_out,
    float* __restrict__ z_g,
    float* __restrict__ commit_partial)
{
    __shared__ float embLds[KCODES * CD];   // 32000 B: full codebook per block
    __shared__ float zLds[8][16 * CD];      // per-wave z tile (C-layout -> A-layout)
    __shared__ float qLds[8][16 * CD];      // per-wave gathered quant tile
    __shared__ int   idxLds[8][16];
    __shared__ float commitLds[8];

    const int tid  = threadIdx.x;
    const int lane = tid & 31;
    const int wave = tid >> 5;
    const int half = lane >> 4;             // 0: lanes 0-15, 1: lanes 16-31
    const int l16  = lane & 15;

    // ---- stage codebook via CDNA5 async copy engine (ASYNCcnt-tracked,
    //      LDS-direct, no VGPR round trip). 2000 x 16B, 256 lanes per pass.
    {
        const uint32_t ldsBase = (uint32_t)(uintptr_t)(&embLds[0]); // low 32b = LDS offset
        #pragma unroll
        for (int it = 0; it < 8; ++it) {
            const int byteOff = it * 4096 + tid * 16;
            if (byteOff < KCODES * CD * 4) {
                const uint32_t ldsOff = ldsBase + (uint32_t)byteOff;
                const uint32_t gOff   = (uint32_t)byteOff;
                asm volatile("global_load_async_to_lds_b128 %0, %1, %2"
                             :: "v"(ldsOff), "v"(gOff), "s"(embed)
                             : "memory");
            }
        }
        asm volatile("s_wait_asynccnt 0" ::: "memory");
    }
    __syncthreads();

    const int tile      = blockIdx.x * 8 + wave;
    const int tokenBase = tile * 16;

    // ---- z tile: [16 tokens x 16 code-dims] = x[16x64] @ W_in[64x16], 16 k-steps
    v8f zc = {};
    #pragma unroll
    for (int s = 0; s < 16; ++s) {
        const int k = 4 * s + 2 * half;
        const float* xp = x + (size_t)(tokenBase + l16) * DIM + k;
        v2f a; a.x = xp[0];              a.y = xp[1];                   // A: M=l16, K=k,k+1
        v2f b; b.x = W_in[k * CD + l16]; b.y = W_in[(k + 1) * CD + l16]; // B: N=l16
        zc = wmma_f32(a, b, zc);
    }
    {
        const float bi = b_in[l16];
        #pragma unroll
        for (int r = 0; r < 8; ++r) {
            const int M = r + 8 * half;          // C-layout row
            const float zv = zc[r] + bi;
            zLds[wave][M * CD + l16] = zv;
            z_g[(size_t)(tokenBase + M) * CD + l16] = zv;
        }
    }
    __syncthreads();

    const float* zt = &zLds[wave][0];

    // ---- hoist z A-fragments: invariant over all 32 code tiles
    v2f za[4];
    #pragma unroll
    for (int s = 0; s < 4; ++s) {
        const int k = 4 * s + 2 * half;
        za[s].x = zt[l16 * CD + k];
        za[s].y = zt[l16 * CD + k + 1];
    }

    // ---- nearest code: maximize 2*z.e - |e|^2 over 500 codes.
    //      Two code tiles per iteration -> two independent WMMA accumulate
    //      chains (breaks the serial RAW on the accumulator).
    float best[8];
    int   bidx[8];
    #pragma unroll
    for (int r = 0; r < 8; ++r) { best[r] = -__builtin_inff(); bidx[r] = 0; }

    for (int t = 0; t < KTILES; t += 2) {
        const int  nA     = t * 16 + l16;
        const int  nB     = nA + 16;
        const bool validA = (nA < KCODES);
        const bool validB = (nB < KCODES);
        v8f   accA = {}, accB = {};
        float nrmA = 0.0f, nrmB = 0.0f;
        #pragma unroll
        for (int s = 0; s < 4; ++s) {
            const int k = 4 * s + 2 * half;
            v2f bA, bB;
            bA.x = validA ? embLds[nA * CD + k]     : 0.0f;
            bA.y = validA ? embLds[nA * CD + k + 1] : 0.0f;
            bB.x = validB ? embLds[nB * CD + k]     : 0.0f;
            bB.y = validB ? embLds[nB * CD + k + 1] : 0.0f;
            nrmA += bA.x * bA.x + bA.y * bA.y;   // partial |e|^2 from B-frag loads
            nrmB += bB.x * bB.x + bB.y * bB.y;
            accA = wmma_f32(za[s], bA, accA);
            accB = wmma_f32(za[s], bB, accB);
        }
        nrmA += __shfl_xor(nrmA, 16, 32);        // other half holds the other 8 dims
        nrmB += __shfl_xor(nrmB, 16, 32);
        #pragma unroll
        for (int r = 0; r < 8; ++r) {            // tile t before t+1: strict > keeps
            const float scA = validA ? (2.0f * accA[r] - nrmA) : -__builtin_inff();
            if (scA > best[r]) { best[r] = scA; bidx[r] = nA; }
            const float scB = validB ? (2.0f * accB[r] - nrmB) : -__builtin_inff();
            if (scB > best[r]) { best[r] = scB; bidx[r] = nB; }
        }
    }

    // reduce argmax across the 16 lanes sharing each row (tie-break: smaller index)
    #pragma unroll
    for (int off = 1; off <= 8; off <<= 1) {
        #pragma unroll
        for (int r = 0; r < 8; ++r) {
            const float os = __shfl_xor(best[r], off, 32);
            const int   oi = __shfl_xor(bidx[r], off, 32);
            if (os > best[r] || (os == best[r] && oi < bidx[r])) {
                best[r] = os; bidx[r] = oi;
            }
        }
    }
    if (l16 == 0) {
        #pragma unroll
        for (int r = 0; r < 8; ++r) idxLds[wave][half * 8 + r] = bidx[r];
    }
    __syncthreads();

    if (half == 0) idx_out[tokenBase + l16] = (float)idxLds[wave][l16];

    // ---- gather quant into LDS + commitment-loss partial (no atomics)
    {
        const int m  = lane >> 1;                 // token within tile
        const int c0 = (lane & 1) * 8;            // 8 dims per lane
        const int qi = idxLds[wave][m];
        float cpart = 0.0f;
        #pragma unroll
        for (int j = 0; j < 8; ++j) {
            const float e  = embLds[qi * CD + c0 + j];
            const float zv = zLds[wave][m * CD + c0 + j];
            qLds[wave][m * CD + c0 + j] = e;
            const float df = e - zv;
            cpart += df * df;
        }
        #pragma unroll
        for (int off = 16; off > 0; off >>= 1) cpart += __shfl_xor(cpart, off, 32);
        if (lane == 0) commitLds[wave] = cpart;
    }
    __syncthreads();
    if (tid == 0) {
        float s = 0.0f;
        #pragma unroll
        for (int w = 0; w < 8; ++w) s += commitLds[w];
        commit_partial[blockIdx.x] = s;
    }

    // ---- out tile: quant[16x16] @ W_out[16x64] + b_out, 4 N-tiles x 4 k-steps
    const float* qt = &qLds[wave][0];
    v2f qa[4];
    #pragma unroll
    for (int s = 0; s < 4; ++s) {
        const int k = 4 * s + 2 * half;
        qa[s].x = qt[l16 * CD + k];
        qa[s].y = qt[l16 * CD + k + 1];
    }
    #pragma unroll
    for (int nt = 0; nt < 4; ++nt) {
        const int n0 = nt * 16;
        v8f oc = {};
        #pragma unroll
        for (int s = 0; s < 4; ++s) {
            const int k = 4 * s + 2 * half;
            v2f b; b.x = W_out[k * DIM + n0 + l16];  b.y = W_out[(k + 1) * DIM + n0 + l16];
            oc = wmma_f32(qa[s], b, oc);
        }
        const float bo = b_out[n0 + l16];
        #pragma unroll
        for (int r = 0; r < 8; ++r) {
            const int M = r + 8 * half;
            out[(size_t)(tokenBase + M) * DIM + n0 + l16] = oc[r] + bo;
        }
    }
}

// ---------------------------------------------------------------------------
// Deterministic segment sums: one block per code; fixed-stride scan + tree.
// ---------------------------------------------------------------------------
__global__ __launch_bounds__(256) void vq_segsum_kernel(
    const float* __restrict__ idx_out,
    const float* __restrict__ z_g,
    float* __restrict__ counts,
    float* __restrict__ embed_sum)
{
    __shared__ float red[256];
    const int code = blockIdx.x;
    const int tid  = threadIdx.x;

    float cnt = 0.0f;
    float sum[CD];
    #pragma unroll
    for (int j = 0; j < CD; ++j) sum[j] = 0.0f;

    for (int tkn = tid; tkn < T_TOKENS; tkn += 256) {
        if ((int)idx_out[tkn] == code) {
            cnt += 1.0f;
            const float* zp = z_g + (size_t)tkn * CD;
            #pragma unroll
            for (int j = 0; j < CD; ++j) sum[j] += zp[j];
        }
    }
    red[tid] = cnt; __syncthreads();
    for (int off = 128; off > 0; off >>= 1) {
        if (tid < off) red[tid] += red[tid + off];
        __syncthreads();
    }
    if (tid == 0) counts[code] = red[0];
    for (int j = 0; j < CD; ++j) {
        __syncthreads();
        red[tid] = sum[j]; __syncthreads();
        for (int off = 128; off > 0; off >>= 1) {
            if (tid < off) red[tid] += red[tid + off];
            __syncthreads();
        }
        if (tid == 0) embed_sum[code * CD + j] = red[0];
    }
}

// ---------------------------------------------------------------------------
// EMA codebook update + Laplace smoothing + loss finalization (single block).
// ---------------------------------------------------------------------------
__global__ __launch_bounds__(512) void vq_update_kernel(
    const float* __restrict__ cluster_size,
    const float* __restrict__ embed_avg,
    const float* __restrict__ counts,
    const float* __restrict__ embed_sum,
    const float* __restrict__ commit_partial,
    float* __restrict__ new_embed,
    float* __restrict__ loss_out)
{
    __shared__ float red[512];
    __shared__ float inv_sm[KCODES];
    const int tid = threadIdx.x;

    float ncs = 0.0f;
    if (tid < KCODES) ncs = cluster_size[tid] * DECAY + ONE_M_D * counts[tid];
    red[tid] = ncs;
    __syncthreads();
    for (int off = 256; off > 0; off >>= 1) {
        if (tid < off) red[tid] += red[tid + off];
        __syncthreads();
    }
    const float n_total = red[0];
    __syncthreads();

    if (tid < KCODES) {
        const float sm = (ncs + EPS_F) / (n_total + (float)KCODES * EPS_F) * n_total;
        inv_sm[tid] = 1.0f / sm;
    }

    red[tid] = commit_partial[tid];   // FUSED_BLOCKS == 512 partials
    __syncthreads();
    for (int off = 256; off > 0; off >>= 1) {
        if (tid < off) red[tid] += red[tid + off];
        __syncthreads();
    }
    if (tid == 0) loss_out[0] = red[0] * (COMMIT_W / (float)((size_t)T_TOKENS * CD));
    __syncthreads();

    for (int i = tid; i < KCODES * CD; i += 512) {
        const float nea = embed_avg[i] * DECAY + ONE_M_D * embed_sum[i];
        new_embed[i] = nea * inv_sm[i >> 4];
    }
}

extern "C" void kernel_launch(void* const* d_in, const int* in_sizes, int n_in,
                              void* d_out, int out_size, void* d_ws, size_t ws_size,
                              hipStream_t stream) {
    const float* x            = (const float*)d_in[0];
    const float* W_in         = (const float*)d_in[1];
    const float* b_in         = (const float*)d_in[2];
    const float* W_out        = (const float*)d_in[3];
    const float* b_out        = (const float*)d_in[4];
    const float* embed        = (const float*)d_in[5];
    const float* cluster_size = (const float*)d_in[6];
    const float* embed_avg    = (const float*)d_in[7];

    // outputs concatenated flat in return order: out, idx, commit_loss, new_embed
    float* out     = (float*)d_out;
    float* idx_out = out + (size_t)T_TOKENS * DIM;
    float* loss    = idx_out + T_TOKENS;
    float* newemb  = loss + 1;

    // workspace: z (4 MB) | commit partials (512) | counts (500) | embed_sum (8000)
    float* ws       = (float*)d_ws;
    float* z_g      = ws;
    float* commit_p = z_g + (size_t)T_TOKENS * CD;
    float* counts   = commit_p + FUSED_BLOCKS;
    float* emb_sum  = counts + KCODES;

    vq_fused_kernel<<<FUSED_BLOCKS, 256, 0, stream>>>(
        x, W_in, b_in, W_out, b_out, embed, out, idx_out, z_g, commit_p);
    vq_segsum_kernel<<<KCODES, 256, 0, stream>>>(idx_out, z_g, counts, emb_sum);
    vq_update_kernel<<<1, 512, 0, stream>>>(
        cluster_size, embed_avg, counts, emb_sum, commit_p, newemb, loss);
}